// EzDetectLoss_68204080661284
// MI455X (gfx1250) — compile-verified
//
#include <hip/hip_runtime.h>
#include <hip/hip_bf16.h>

#define Bb   64
#define Nn   8732
#define Cc   21
#define Mm   16
#define TOT  (Bb * Nn)          // 558848 = 16 * 34928, divides evenly into 16-row WMMA tiles

typedef __attribute__((ext_vector_type(2))) float v2f;
typedef __attribute__((ext_vector_type(8))) float v8f;

__device__ __forceinline__ unsigned f2key(float f) {
    unsigned u = __float_as_uint(f);
    return (u & 0x80000000u) ? ~u : (u | 0x80000000u);   // ascending float <-> ascending uint
}
__device__ __forceinline__ float key2f(unsigned k) {
    unsigned u = (k & 0x80000000u) ? (k & 0x7FFFFFFFu) : ~k;
    return __uint_as_float(u);
}

// ---------------------------------------------------------------------------
// K1: per-row log-sum-exp via V_WMMA_F32_16X16X4_F32 (B = ones => exact f32 row
// sums of exp terms), write sortable key of bg = conf[...,0] - lse.
// One wave handles 16 rows; 8 waves / block; TOT/128 blocks, no partial tiles.
// ---------------------------------------------------------------------------
__global__ __launch_bounds__(256) void k_softmax_keys(const float* __restrict__ conf,
                                                      unsigned* __restrict__ keys) {
    const int lane    = threadIdx.x & 31;
    const int wave    = threadIdx.x >> 5;
    const int tile    = blockIdx.x * 8 + wave;
    const int rowBase = tile * 16;
    const int myRow   = rowBase + (lane & 15);
    const int half    = lane >> 4;                 // A-matrix K-half per ISA layout
    const float* rp   = conf + (size_t)myRow * Cc;

    // Row max: half 0 covers c=0..10, half 1 covers c=11..20; combine across halves.
    float mx = -3.402823466e38f;
    for (int i = 0; i < 11; ++i) {
        int c = half * 11 + i;
        if (c < Cc) mx = fmaxf(mx, rp[c]);
    }
    mx = fmaxf(mx, __shfl_xor(mx, 16, 32));        // both halves now hold full row max

    // Sum of exp via 6 chained WMMAs: A[m][k] = exp(x - mx) (zero-padded to K=24),
    // B = ones(4x16)  =>  D[m][n] = sum_k A[m][k]  (replicated over n).
    v2f ones; ones.x = 1.0f; ones.y = 1.0f;
    v8f acc = {0.0f, 0.0f, 0.0f, 0.0f, 0.0f, 0.0f, 0.0f, 0.0f};
    for (int kk = 0; kk < 6; ++kk) {
        int c0 = kk * 4 + half * 2;                // lane<16: K=0,1 ; lane>=16: K=2,3
        int c1 = c0 + 1;
        v2f a;
        a.x = (c0 < Cc) ? expf(rp[c0] - mx) : 0.0f;
        a.y = (c1 < Cc) ? expf(rp[c1] - mx) : 0.0f;
        acc = __builtin_amdgcn_wmma_f32_16x16x4_f32(false, a, false, ones,
                                                    (short)0, acc, false, false);
    }

    // D layout: VGPR j -> M=j (lanes 0-15), M=8+j (lanes 16-31).
    // Writers: lanes 0..7 emit rows 0..7 (slot=lane), lanes 16..23 emit rows 8..15
    // (slot=lane-16). Row max for row r lives in lane r -> shuffle for lanes>=16.
    int j = lane & 7;
    float s = (j == 0) ? acc[0] : (j == 1) ? acc[1] : (j == 2) ? acc[2] :
              (j == 3) ? acc[3] : (j == 4) ? acc[4] : (j == 5) ? acc[5] :
              (j == 6) ? acc[6] : acc[7];
    int   src = (lane < 16) ? lane : (lane - 8);
    float mxr = __shfl(mx, src, 32);
    if ((lane & 15) < 8) {
        int r    = (lane & 7) + ((lane >> 4) << 3);
        int orow = rowBase + r;
        float lse = mxr + logf(s);
        float bg  = conf[(size_t)orow * Cc] - lse; // log-softmax of class 0
        keys[orow] = f2key(bg);
    }
}

// ---------------------------------------------------------------------------
// K2: parse targets (B*M = 1024 threads, single block). Overwrites assigned
// rows' keys with key(0.0); computes bbox smooth-L1 sum, positive CE sum,
// pnum, validCount via deterministic shared-memory tree reductions.
// state[]: 0=pnum 1=validCount 2=prefix 3=need 4=Ktot 5=pos_ce(bits) 6=sl1(bits)
// ---------------------------------------------------------------------------
__global__ __launch_bounds__(1024) void k_targets(const float* __restrict__ conf,
                                                  const float* __restrict__ bbox,
                                                  const float* __restrict__ target,
                                                  const float* __restrict__ pred,
                                                  unsigned* __restrict__ keys,
                                                  unsigned* __restrict__ state) {
    __shared__ float sf[1024];
    __shared__ int   si[1024];
    const int t = threadIdx.x;
    const int b = t >> 4, m = t & 15;

    const float* tg = target + (size_t)b * (1 + 6 * Mm);
    const int num   = (int)tg[0];
    const float* e  = tg + 1 + m * 6;
    const int cls   = (int)e[0];
    const float tx1 = e[1], ty1 = e[2], tx2 = e[3], ty2 = e[4];
    const int k     = (int)e[5];
    const bool valid = (m < num);
    const bool pos   = valid && (cls > 0);

    float sl1 = 0.0f, ce = 0.0f;
    if (valid) {
        const size_t row = (size_t)b * Nn + k;
        keys[row] = 0x80000000u;                   // assigned => bg := 0.0
        const float* pb = pred + (size_t)k * 4;
        float pcx = (pb[0] + pb[2]) * 0.5f, pcy = (pb[1] + pb[3]) * 0.5f;
        float pw  = pb[2] - pb[0],          ph  = pb[3] - pb[1];
        float tcx = (tx1 + tx2) * 0.5f,     tcy = (ty1 + ty2) * 0.5f;
        float tw  = tx2 - tx1,              th  = ty2 - ty1;
        float eb[4] = { (tcx - pcx) / pw, (tcy - pcy) / ph, logf(tw / pw), logf(th / ph) };
        const float* bbp = bbox + row * 4;
        for (int i = 0; i < 4; ++i) {
            float d = bbp[i] - eb[i];
            float ad = fabsf(d);
            sl1 += (ad < 1.0f) ? 0.5f * d * d : (ad - 0.5f);
        }
        if (pos) {
            const float* cp = conf + row * Cc;
            float mx = cp[0];
            for (int c = 1; c < Cc; ++c) mx = fmaxf(mx, cp[c]);
            float s = 0.0f;
            for (int c = 0; c < Cc; ++c) s += expf(cp[c] - mx);
            ce = (mx + logf(s)) - cp[cls];         // -log_softmax at cls
        }
    }

    // deterministic tree reductions
    sf[t] = sl1; __syncthreads();
    for (int o = 512; o > 0; o >>= 1) { if (t < o) sf[t] += sf[t + o]; __syncthreads(); }
    if (t == 0) state[6] = __float_as_uint(sf[0]);
    __syncthreads();
    sf[t] = ce; __syncthreads();
    for (int o = 512; o > 0; o >>= 1) { if (t < o) sf[t] += sf[t + o]; __syncthreads(); }
    if (t == 0) state[5] = __float_as_uint(sf[0]);
    __syncthreads();
    si[t] = valid ? 1 : 0; __syncthreads();
    for (int o = 512; o > 0; o >>= 1) { if (t < o) si[t] += si[t + o]; __syncthreads(); }
    if (t == 0) state[1] = (unsigned)si[0];
    __syncthreads();
    si[t] = pos ? 1 : 0; __syncthreads();
    for (int o = 512; o > 0; o >>= 1) { if (t < o) si[t] += si[t + o]; __syncthreads(); }
    if (t == 0) state[0] = (unsigned)si[0];
}

// K3: init radix-select state (need = K = 3*pnum) and zero histogram.
__global__ __launch_bounds__(256) void k_select_init(unsigned* __restrict__ state,
                                                     int* __restrict__ hist) {
    hist[threadIdx.x] = 0;
    if (threadIdx.x == 0) {
        unsigned K = 3u * state[0];
        state[2] = 0u;   // prefix
        state[3] = K;    // need
        state[4] = K;    // total K
    }
}

// K4: 256-bin histogram of keys whose already-fixed high bits match prefix.
__global__ __launch_bounds__(256) void k_hist(const unsigned* __restrict__ keys,
                                              const unsigned* __restrict__ state,
                                              int* __restrict__ hist, int shift) {
    const unsigned prefix = state[2];
    int i      = blockIdx.x * blockDim.x + threadIdx.x;
    int stride = gridDim.x * blockDim.x;
    for (; i < TOT; i += stride) {
        unsigned key = keys[i];
        if ((((key ^ prefix) >> shift) >> 8) == 0u)   // high bits match (works for shift=24 too)
            atomicAdd(&hist[(key >> shift) & 0xFFu], 1);
    }
}

// K5: pick the bucket containing the K-th smallest; refine prefix/need; rezero hist.
__global__ __launch_bounds__(256) void k_pick(unsigned* __restrict__ state,
                                              int* __restrict__ hist, int shift) {
    __shared__ int h[256];
    h[threadIdx.x]    = hist[threadIdx.x];
    hist[threadIdx.x] = 0;
    __syncthreads();
    if (threadIdx.x == 0) {
        unsigned need = state[3];
        int bin = 0;
        for (; bin < 255; ++bin) {
            unsigned c = (unsigned)h[bin];
            if (need > c) need -= c; else break;
        }
        state[2] |= ((unsigned)bin) << shift;
        state[3]  = need;
    }
}

// K6: sum of (-bg) over keys strictly below threshold T; deterministic per-block partials.
__global__ __launch_bounds__(256) void k_negsum(const unsigned* __restrict__ keys,
                                                const unsigned* __restrict__ state,
                                                float* __restrict__ blockSums) {
    __shared__ float sf[256];
    const unsigned T = state[2];
    float local = 0.0f;
    int i      = blockIdx.x * blockDim.x + threadIdx.x;
    int stride = gridDim.x * blockDim.x;
    for (; i < TOT; i += stride) {
        unsigned key = keys[i];
        if (key < T) local += -key2f(key);
    }
    sf[threadIdx.x] = local; __syncthreads();
    for (int o = 128; o > 0; o >>= 1) { if (threadIdx.x < o) sf[threadIdx.x] += sf[threadIdx.x + o]; __syncthreads(); }
    if (threadIdx.x == 0) blockSums[blockIdx.x] = sf[0];
}

// K7: final fixed-order reduction of 512 partials; add tie contribution
// (ties share an identical float value); write [conf_loss, bbox_loss].
__global__ __launch_bounds__(256) void k_finalize(const unsigned* __restrict__ state,
                                                  const float* __restrict__ blockSums,
                                                  float* __restrict__ out) {
    __shared__ float sf[256];
    sf[threadIdx.x] = blockSums[threadIdx.x] + blockSums[threadIdx.x + 256];
    __syncthreads();
    for (int o = 128; o > 0; o >>= 1) { if (threadIdx.x < o) sf[threadIdx.x] += sf[threadIdx.x + o]; __syncthreads(); }
    if (threadIdx.x == 0) {
        float negsum     = sf[0];
        unsigned T       = state[2];
        unsigned need    = state[3];
        unsigned K       = state[4];
        unsigned pnum    = state[0];
        unsigned validCt = state[1];
        if (need > 0u) negsum += (float)need * (-key2f(T));
        float pos_ce = __uint_as_float(state[5]);
        float sl1    = __uint_as_float(state[6]);
        unsigned selcount = pnum + K;              // posmask | negmask (disjoint)
        if (selcount < 1u) selcount = 1u;
        unsigned bden = 4u * validCt;
        if (bden < 1u) bden = 1u;
        out[0] = (pos_ce + negsum) / (float)selcount;
        out[1] = sl1 / (float)bden;
    }
}

extern "C" void kernel_launch(void* const* d_in, const int* in_sizes, int n_in,
                              void* d_out, int out_size, void* d_ws, size_t ws_size,
                              hipStream_t stream) {
    const float* confOut = (const float*)d_in[0];   // (64, 8732, 21)
    const float* bboxOut = (const float*)d_in[1];   // (64, 8732, 4)
    const float* target  = (const float*)d_in[2];   // (64, 97)
    const float* pred    = (const float*)d_in[3];   // (8732, 4)
    float* out = (float*)d_out;                     // [conf_loss, bbox_loss]

    char* ws = (char*)d_ws;
    unsigned* keys      = (unsigned*)ws;                               // TOT u32
    int*      hist      = (int*)     (ws + (size_t)TOT * 4);           // 256 i32
    unsigned* state     = (unsigned*)(ws + (size_t)TOT * 4 + 1024);    // 16 u32
    float*    blockSums = (float*)   (ws + (size_t)TOT * 4 + 1024 + 64); // 512 f32

    k_softmax_keys<<<TOT / 128, 256, 0, stream>>>(confOut, keys);
    k_targets<<<1, 1024, 0, stream>>>(confOut, bboxOut, target, pred, keys, state);
    k_select_init<<<1, 256, 0, stream>>>(state, hist);
    for (int p = 0; p < 4; ++p) {
        int shift = 24 - 8 * p;
        k_hist<<<512, 256, 0, stream>>>(keys, state, hist, shift);
        k_pick<<<1, 256, 0, stream>>>(state, hist, shift);
    }
    k_negsum<<<512, 256, 0, stream>>>(keys, state, blockSums);
    k_finalize<<<1, 256, 0, stream>>>(state, blockSums, out);
}